// Channel_9388798509339
// MI455X (gfx1250) — compile-verified
//
#include <hip/hip_runtime.h>
#include <math.h>

// ---------------- problem constants (from reference) ----------------
#define SMK   567              // (S+1)*(M+N_PILOT+K) = 7*81
#define LTAP  8                // channel taps
#define TOUT  (SMK + LTAP - 1) // 574
#define MFREQ 64               // FFT length
#define NROWS 16384            // N*P = 4096*4

typedef __attribute__((ext_vector_type(2))) float v2f;
typedef __attribute__((ext_vector_type(8))) float v8f;

#if __has_builtin(__builtin_amdgcn_tensor_load_to_lds) && \
    __has_builtin(__builtin_amdgcn_s_wait_tensorcnt)
#define USE_TDM 1
typedef __attribute__((ext_vector_type(4))) unsigned int v4u;
typedef __attribute__((ext_vector_type(8))) int          v8i;
typedef __attribute__((ext_vector_type(4))) int          v4i;
#else
#define USE_TDM 0
#endif

// =====================================================================
// Kernel 1: complex FIR full convolution, one block per batch row.
// x row (1134 floats = 4536 B) staged into LDS by the Tensor Data Mover
// (one TDM descriptor per block, all fields wave-uniform), zero-padded
// both sides so the tap loop is branch-free. Taps live in VGPRs.
// =====================================================================
__global__ __launch_bounds__(256) void fir_conv_kernel(
    const float* __restrict__ x, const float* __restrict__ cof,
    float* __restrict__ out)
{
  const int row = blockIdx.x;
  const int tid = threadIdx.x;

  // 7 zero pad + 567 samples + 7 zero pad  (index with (i - l + 7))
  __shared__ float2 sx[TOUT + LTAP - 1];  // 581 float2 = 4648 B

#if USE_TDM
  if (tid < 32) {  // wave 0 issues the DMA (TDM ignores EXEC; one issue/wave)
    const unsigned lds_off =
        (unsigned)(uintptr_t)(void*)sx + 7u * sizeof(float2); // dest = sx[7]
    const unsigned long long ga =
        (unsigned long long)(uintptr_t)(x + (size_t)row * (2 * SMK));
    const unsigned elems = 2 * SMK;  // 1134 dwords

    // ---- D# group 0: count=1 | lds_addr | global_addr | type=2 ----
    v4u g0;
    g0.x = 1u;                                   // count=1, user desc
    g0.y = lds_off;                              // lds_addr [63:32]
    g0.z = (unsigned)(ga & 0xFFFFFFFFu);         // global_addr [95:64]
    g0.w = (unsigned)((ga >> 32) & 0x01FFFFFFu)  // global_addr [120:96]
           | (2u << 30);                         // type=2 ("image") [127:126]

    // ---- D# group 1: 1-line tile of 1134 4-byte elements ----
    v8i g1;
    g1[0] = (int)(2u << 16);          // data_size=2 (4B); mask/flags/pad = 0
    g1[1] = (int)(elems << 16);       // tensor_dim0[15:0] @ [63:48]
    g1[2] = (int)(1u << 16);          // tensor_dim0[31:16]=0 | tensor_dim1=1
    g1[3] = (int)(elems << 16);       // tensor_dim1[31:16]=0 | tile_dim0=1134
    g1[4] = 1;                        // tile_dim1=1, tile_dim2=0
    g1[5] = (int)elems;               // tensor_dim0_stride[31:0]
    g1[6] = (int)(elems << 16);       // stride0[47:32]=0 | tensor_dim1_stride lo
    g1[7] = 0;

    v4i gz  = {0, 0, 0, 0};                      // groups 2/3 unused (<=2D)
    v8i gz8 = {0, 0, 0, 0, 0, 0, 0, 0};          // trailing group (unused)
    // 6-arg form (clang-23 / therock-10.0 toolchain)
    __builtin_amdgcn_tensor_load_to_lds(g0, g1, gz, gz, gz8, 0);
  }
  // zero pads (outside the TDM destination range) with normal DS stores
  if (tid < 7) {
    sx[tid]        = make_float2(0.f, 0.f);
    sx[TOUT + tid] = make_float2(0.f, 0.f);
  }
#else
  const float2* __restrict__ xin =
      reinterpret_cast<const float2*>(x) + (size_t)row * SMK;
  for (int j = tid; j < SMK; j += 256)
    sx[j + 7] = xin[j];
  if (tid < 7) {
    sx[tid]        = make_float2(0.f, 0.f);
    sx[TOUT + tid] = make_float2(0.f, 0.f);
  }
#endif

  // taps: all lanes read the same 64 bytes -> single cacheline broadcast
  float hr[LTAP], hi[LTAP];
  const float2* __restrict__ hc =
      reinterpret_cast<const float2*>(cof) + (size_t)row * LTAP;
#pragma unroll
  for (int l = 0; l < LTAP; ++l) { float2 t = hc[l]; hr[l] = t.x; hi[l] = t.y; }

#if USE_TDM
  if (tid < 32) __builtin_amdgcn_s_wait_tensorcnt(0);  // wave 0: DMA done
#endif
  __syncthreads();

  float2* __restrict__ o = reinterpret_cast<float2*>(out) + (size_t)row * TOUT;
  for (int i = tid; i < TOUT; i += 256) {
    float ar = 0.f, ai = 0.f;
#pragma unroll
    for (int l = 0; l < LTAP; ++l) {
      float2 v = sx[i - l + 7];
      ar = fmaf(hr[l],  v.x, ar);
      ar = fmaf(-hi[l], v.y, ar);
      ai = fmaf(hr[l],  v.y, ai);
      ai = fmaf(hi[l],  v.x, ai);
    }
    o[i] = make_float2(ar, ai);
  }
}

// =====================================================================
// Kernel 2: H_true = taps @ realified-DFT(16x128) via chained
// V_WMMA_F32_16X16X4_F32. One wave handles 16 batch rows.
// DFT weights generated in-register with hardware v_sin/v_cos:
// phase is an exact integer /64, i.e. turn units -> no range reduction.
// Key simplification: for kk = 4c+v+2h, tap index l = 2c+h for BOTH
// B components and kk&1 == v, so one sin/cos pair per (tile, chunk).
// =====================================================================
__global__ __launch_bounds__(256) void dft_wmma_kernel(
    const float* __restrict__ cof, float* __restrict__ Hout)
{
  const int lane  = threadIdx.x & 31;
  const int wave  = threadIdx.x >> 5;
  const int gwave = blockIdx.x * 8 + wave;
  const int row0  = gwave * 16;

  const int m = lane & 15;   // M (and N) position within tile
  const int h = lane >> 4;   // lane-half selector

  // ---- A-matrix: taps of 16 rows. K index kk = c*4 + v + 2*h ----
  const float* __restrict__ arow = cof + (size_t)(row0 + m) * (2 * LTAP);
  v2f A[4];
#pragma unroll
  for (int c = 0; c < 4; ++c) {
    A[c].x = arow[c * 4 + 0 + 2 * h];
    A[c].y = arow[c * 4 + 1 + 2 * h];
  }

  const int  floc = m >> 1;        // local frequency within 8-freq tile
  const bool d    = (m & 1) != 0;  // false -> real col, true -> imag col
  const float inv64 = 1.0f / (float)MFREQ;

#pragma unroll
  for (int t = 0; t < 8; ++t) {      // 8 N-tiles cover 64 complex freqs
    v8f acc = {};
#pragma unroll
    for (int c = 0; c < 4; ++c) {    // chain K = 16 as 4 x (K=4)
      const int l  = 2 * c + h;                      // tap index for kk and kk+1
      const int f  = 8 * t + floc;                   // frequency bin
      const int ph = (f * l) & (MFREQ - 1);          // phase mod 64
      const float u  = (float)ph * inv64;            // turns in [0,1)
      const float sn = __builtin_amdgcn_sinf(u);     // v_sin_f32
      const float cs = __builtin_amdgcn_cosf(u);     // v_cos_f32
      v2f B;
      B.x = d ? -sn : cs;   // W[2l+0, n]:  cos | -sin
      B.y = d ?  cs : sn;   // W[2l+1, n]:  sin |  cos
      acc = __builtin_amdgcn_wmma_f32_16x16x4_f32(
          /*neg_a=*/false, A[c], /*neg_b=*/false, B,
          /*c_mod=*/(short)0, acc, /*reuse_a=*/false, /*reuse_b=*/false);
    }
    // ---- D store: VGPR j -> M = j + 8*h, N = m ----
#pragma unroll
    for (int j = 0; j < 8; ++j) {
      const int M = j + 8 * h;
      Hout[(size_t)(row0 + M) * (2 * MFREQ) + 16 * t + m] = acc[j];
    }
  }
}

// =====================================================================
extern "C" void kernel_launch(void* const* d_in, const int* in_sizes, int n_in,
                              void* d_out, int out_size, void* d_ws, size_t ws_size,
                              hipStream_t stream)
{
  (void)in_sizes; (void)n_in; (void)d_ws; (void)ws_size; (void)out_size;

  const float* x   = (const float*)d_in[0];   // (N,P,SMK,2) f32
  const float* cof = (const float*)d_in[1];   // (N,P,L,2)   f32
  // d_in[2] is M (=64), known at compile time.

  float* out  = (float*)d_out;                         // (N,P,574,2)
  float* Hout = out + (size_t)NROWS * TOUT * 2;        // (N,P,64,2)

  fir_conv_kernel<<<NROWS, 256, 0, stream>>>(x, cof, out);

  // NROWS/16 = 1024 waves, 8 waves per 256-thread block -> 128 blocks
  dft_wmma_kernel<<<NROWS / (16 * 8), 256, 0, stream>>>(cof, Hout);
}